// ViterbiLoss_14156212207716
// MI455X (gfx1250) — compile-verified
//
#include <hip/hip_runtime.h>

#define KTAGS     64
#define START_TAG 62
#define END_TAG   63
#define DEPTH     8          // LDS tile pipeline depth (8 x 16KB = 128KB)
#define TILE_ELEM (KTAGS * KTAGS)   // 4096 floats = 16KB
#define BLOCK     1024

// ---------------- CDNA5 async global->LDS support probes ----------------
#if defined(__has_builtin)
#  if __has_builtin(__builtin_amdgcn_global_load_async_to_lds_b128)
#    define HAVE_ASYNC_LDS 1
#  endif
#endif
#ifndef HAVE_ASYNC_LDS
#  define HAVE_ASYNC_LDS 0
#endif

#if defined(__has_builtin)
#  if __has_builtin(__builtin_amdgcn_s_wait_asynccnt)
#    define WAIT_ASYNC(n) __builtin_amdgcn_s_wait_asynccnt(n)
#  endif
#endif
#ifndef WAIT_ASYNC
#  if HAVE_ASYNC_LDS
#    define WAIT_ASYNC(n) asm volatile("s_wait_asynccnt %0" ::"n"(n) : "memory")
#  else
#    define WAIT_ASYNC(n) ((void)0)
#  endif
#endif

// Exact builtin parameter types: (v4i AS1* src, v4i AS3* dst, imm, imm)
typedef int v4i_t __attribute__((ext_vector_type(4)));
typedef __attribute__((address_space(1))) v4i_t gv4i_t;  // global
typedef __attribute__((address_space(3))) v4i_t lv4i_t;  // LDS

// Copy 16 bytes global -> LDS. Async path uses GLOBAL_LOAD_ASYNC_TO_LDS_B128
// (tracked by ASYNCcnt, per-wave in-order completion); fallback is a
// synchronous register-staged copy (covered by the same barriers).
__device__ __forceinline__ void copy16_g2l(void* lds_dst, const void* g_src) {
#if HAVE_ASYNC_LDS
  __builtin_amdgcn_global_load_async_to_lds_b128(
      (gv4i_t*)g_src, (lv4i_t*)lds_dst, /*imm offset*/ 0, /*cpol*/ 0);
#else
  *reinterpret_cast<float4*>(lds_dst) = *reinterpret_cast<const float4*>(g_src);
#endif
}

// Wait until at most `n` of this wave's async tile-copies are outstanding.
// The builtin needs an immediate, so expand the (block-uniform) tail cases.
__device__ __forceinline__ void wait_async_upto(int n) {
  switch (n) {
    case 0: WAIT_ASYNC(0); break;
    case 1: WAIT_ASYNC(1); break;
    case 2: WAIT_ASYNC(2); break;
    case 3: WAIT_ASYNC(3); break;
    case 4: WAIT_ASYNC(4); break;
    case 5: WAIT_ASYNC(5); break;
    case 6: WAIT_ASYNC(6); break;
    default: WAIT_ASYNC(DEPTH - 1); break;
  }
}

// Reduce max over the 64 values held by tid<64 (two waves) and park the two
// per-wave maxima in sWmax[0..1]; every thread recombines them next step.
__device__ __forceinline__ void wavepair_max_store(float v, int tid,
                                                   float* sWmax) {
  v = fmaxf(v, __shfl_xor(v, 16, 32));
  v = fmaxf(v, __shfl_xor(v, 8, 32));
  v = fmaxf(v, __shfl_xor(v, 4, 32));
  v = fmaxf(v, __shfl_xor(v, 2, 32));
  v = fmaxf(v, __shfl_xor(v, 1, 32));
  if ((tid & 31) == 0) sWmax[tid >> 5] = v;
}

__global__ __launch_bounds__(BLOCK) void crf_forward_kernel(
    const float* __restrict__ scores, const int* __restrict__ targets,
    const int* __restrict__ lengths, float* __restrict__ ws, int T) {
  __shared__ float buf[DEPTH][TILE_ELEM];  // 128 KB tile pipeline
  __shared__ float sPart[BLOCK];           // 4 KB partial sums / reductions
  __shared__ float sAlpha[KTAGS];
  __shared__ float sWmax[2];

  const int tid = threadIdx.x;
  const int b   = blockIdx.x;
  const int len = lengths[b];  // block-uniform scalar load
  const float* base = scores + (size_t)b * (size_t)T * TILE_ELEM;

  // -------------------- gold path score (gather + block sum) ---------------
  float g = 0.f;
  for (int t = tid; t < T; t += BLOCK)
    if (t < len)
      g += base[(size_t)t * TILE_ELEM + targets[(size_t)b * T + t]];
  sPart[tid] = g;
  __syncthreads();
#pragma unroll
  for (int off = BLOCK / 2; off > 0; off >>= 1) {
    if (tid < off) sPart[tid] += sPart[tid + off];
    __syncthreads();
  }
  if (tid == 0) ws[b] = sPart[0];
  if (tid < KTAGS) sAlpha[tid] = 0.f;
  if (tid < 2) sWmax[tid] = 0.f;
  __syncthreads();

  // -------------------- forward recursion with async LDS pipeline ----------
  const int j = tid & 63;   // column (lanes -> consecutive LDS words)
  const int r = tid >> 6;   // row-group 0..15, uniform within a wave

  // Prologue: prefetch tiles 0..DEPTH-1. Each thread moves one 16B chunk,
  // so each wave issues exactly one async instruction per tile.
#pragma unroll
  for (int k = 0; k < DEPTH; ++k)
    if (k < len)
      copy16_g2l((char*)buf[k] + (size_t)tid * 16,
                 (const char*)(base + (size_t)k * TILE_ELEM) + (size_t)tid * 16);

  for (int t = 0; t < len; ++t) {
    // Per-wave in-order completion: after issuing tiles 0..min(t+DEPTH,len)-1,
    // allowing min(DEPTH-1, len-1-t) outstanding guarantees tile t landed.
    const int pend = len - 1 - t;
    wait_async_upto(pend < (DEPTH - 1) ? pend : (DEPTH - 1));
    __syncthreads();  // other waves' chunks of tile t are also done

    const float* tile = buf[t % DEPTH];
    if (t == 0) {
      // alpha = sc0[START_TAG, :]
      if (tid < KTAGS) {
        const float na = tile[START_TAG * KTAGS + tid];
        sAlpha[tid] = na;
        wavepair_max_store(na, tid, sWmax);
      }
      __syncthreads();
    } else {
      // new_alpha[j] = ma + log( sum_i exp(sc[i,j] + alpha[i] - ma) ),
      // ma = max_i alpha[i] (exponent bounded: |sc| small for this workload).
      const float ma = fmaxf(sWmax[0], sWmax[1]);
      float p = 0.f;
#pragma unroll
      for (int s = 0; s < 4; ++s) {
        const int i = r * 4 + s;                       // wave-uniform row
        p += __expf(tile[i * KTAGS + j] + (sAlpha[i] - ma));
      }
      sPart[r * KTAGS + j] = p;  // conflict-free
      __syncthreads();
      if (tid < KTAGS) {
        float ssum = 0.f;
#pragma unroll
        for (int r2 = 0; r2 < 16; ++r2) ssum += sPart[r2 * KTAGS + tid];
        const float na = ma + __logf(ssum);
        sAlpha[tid] = na;
        wavepair_max_store(na, tid, sWmax);
      }
      __syncthreads();
    }

    // Refill the slot just consumed with tile t+DEPTH (all reads of
    // buf[t % DEPTH] finished at the barrier above).
    if (t + DEPTH < len)
      copy16_g2l((char*)buf[t % DEPTH] + (size_t)tid * 16,
                 (const char*)(base + (size_t)(t + DEPTH) * TILE_ELEM) +
                     (size_t)tid * 16);
  }

  if (tid == 0) ws[gridDim.x + b] = sAlpha[END_TAG];
}

__global__ void crf_finalize_kernel(const float* __restrict__ ws,
                                    float* __restrict__ out, int Bn) {
  if (threadIdx.x == 0 && blockIdx.x == 0) {
    float sa = 0.f, sg = 0.f;
    for (int b = 0; b < Bn; ++b) {  // fixed order -> deterministic
      sa += ws[Bn + b];
      sg += ws[b];
    }
    out[0] = (sa - sg) / (float)Bn;
  }
}

extern "C" void kernel_launch(void* const* d_in, const int* in_sizes, int n_in,
                              void* d_out, int out_size, void* d_ws,
                              size_t ws_size, hipStream_t stream) {
  (void)n_in;
  (void)out_size;
  (void)ws_size;
  const float* scores  = (const float*)d_in[0];
  const int*   targets = (const int*)d_in[1];
  const int*   lengths = (const int*)d_in[2];
  const int Bn = in_sizes[2];
  const int Tn = in_sizes[1] / Bn;  // targets is B*T
  float* ws = (float*)d_ws;         // [0..B): gold_b, [B..2B): alpha_end_b

  crf_forward_kernel<<<Bn, BLOCK, 0, stream>>>(scores, targets, lengths, ws,
                                               Tn);
  crf_finalize_kernel<<<1, 64, 0, stream>>>(ws, (float*)d_out, Bn);
}